// TemporalFusionTransformer_20246475833944
// MI455X (gfx1250) — compile-verified
//
#include <hip/hip_runtime.h>
#include <hip/hip_bf16.h>
#include <math.h>

// ---------------------------------------------------------------------------
// MI455X / gfx1250:
//  * f32 tensor path: V_WMMA_F32_16X16X4_F32 (A 16x4 = v2f, B 4x16 = v2f,
//    C/D 16x16 = v8f).  Lane&15 = M row (A) / N col (B); half-wave = K half.
//  * Recurrent scans: weights are loop-invariant across time -> LICM hoisted
//    them into ~2K floats/lane and spilled to scratch last round.  We block
//    that with an opaque per-step offset so the loads stay in the loop and
//    only the accumulators (192/256 VGPRs, covered by the 1024-VGPR file)
//    are live across the step.
//  * GRU weights (192 KB/feature) live in LDS (320 KB/WGP on CDNA5) with a
//    132-float pitch => ds_load_b64 B-fragments are bank-conflict-free.
//  * LSTM weights are streamed from L2, pre-transposed to k-major so the
//    16-lane fragment reads are coalesced.
// ---------------------------------------------------------------------------

typedef __attribute__((ext_vector_type(2))) float v2f;
typedef __attribute__((ext_vector_type(8))) float v8f;

#define WMMA_F32(a, b, c) \
  __builtin_amdgcn_wmma_f32_16x16x4_f32(false, (a), false, (b), (short)0, (c), false, false)

__device__ __forceinline__ float sigmoidf_(float x) {
  return 1.0f / (1.0f + __expf(-x));
}

// Opaque zero: new SSA value every call -> addresses derived from it are not
// loop-invariant -> compiler cannot hoist weight loads out of the time loop.
__device__ __forceinline__ int opaque_zero() {
  int v = 0;
  asm volatile("" : "+s"(v));
  return v;
}

#define HPITCH 132  // LDS row pitch (floats): 132 % 64 == 4 -> conflict-free b64

// ---------------------------------------------------------------------------
// Kernel 1: per-feature GRU scan.
// grid = 256 (feature f = bid>>1, batch-half = bid&1), block = 128 (4 waves).
// Each wave owns 16 batch rows end-to-end.  Whh[f] staged in LDS once.
// ---------------------------------------------------------------------------
__global__ __launch_bounds__(128) void gru_featurewise_kernel(
    const float* __restrict__ x,     // [128,256,128]
    const float* __restrict__ Wih,   // [128,384]
    const float* __restrict__ Whh,   // [128,384,128]
    const float* __restrict__ bih,   // [128,384]
    const float* __restrict__ bhh,   // [128,384]
    float* __restrict__ tc)          // out [B=128, F=128, H=128]
{
  __shared__ float sh_B[384 * HPITCH];     // ~198 KB  Whh[f], row-major [n][k]
  __shared__ float sh_h[64][HPITCH];       // ~33.8 KB hidden state (this half)
  __shared__ float sh_wih[384], sh_bih[384], sh_bhh[384];
  __shared__ float sh_x[64];

  const int f     = blockIdx.x >> 1;
  const int rbase = (blockIdx.x & 1) * 64;  // batch rows [rbase, rbase+64)
  const int tid   = threadIdx.x;
  const int w     = tid >> 5;
  const int l     = tid & 31;
  const int half  = (l < 16) ? 0 : 1;
  const int l15   = l & 15;

  const float* Bg = Whh + (size_t)f * 384 * 128;   // B[k][n] = Bg[n*128+k]
  for (int i = tid; i < 384 * 128; i += 128)
    sh_B[(i >> 7) * HPITCH + (i & 127)] = Bg[i];
  for (int i = tid; i < 64 * 128; i += 128) sh_h[i >> 7][i & 127] = 0.0f;
  for (int i = tid; i < 384; i += 128) {
    sh_wih[i] = Wih[f * 384 + i];
    sh_bih[i] = bih[f * 384 + i];
    sh_bhh[i] = bhh[f * 384 + i];
  }
  __syncthreads();

  const int mrow = w * 16 + l15;   // local A row for this lane

  for (int t = 0; t < 256; ++t) {
    if (tid < 64) sh_x[tid] = x[((size_t)(rbase + tid) * 256 + t) * 128 + f];
    __syncthreads();

    const int zo = opaque_zero();  // blocks cross-step hoisting of B loads

    v8f acc[24];
#pragma unroll
    for (int nt = 0; nt < 24; ++nt) acc[nt] = {};

    for (int k0 = 0; k0 < 128; k0 += 4) {
      const int kk = k0 + 2 * half;
      const v2f a = *(const v2f*)(&sh_h[mrow][kk]);
#pragma unroll
      for (int nt = 0; nt < 24; ++nt) {
        const v2f b = *(const v2f*)(&sh_B[(nt * 16 + l15) * HPITCH + kk + zo]);
        acc[nt] = WMMA_F32(a, b, acc[nt]);
      }
    }

    // Gates directly out of accumulator layout; wave owns its 16 rows.
#pragma unroll
    for (int jt = 0; jt < 8; ++jt) {
#pragma unroll
      for (int v = 0; v < 8; ++v) {
        const int brow = w * 16 + v + 8 * half;   // local row
        const int j    = jt * 16 + l15;
        const float xv = sh_x[brow];
        const float hr = acc[jt][v]      + sh_bhh[j];
        const float hz = acc[8 + jt][v]  + sh_bhh[128 + j];
        const float hn = acc[16 + jt][v] + sh_bhh[256 + j];
        const float ir = xv * sh_wih[j]       + sh_bih[j];
        const float iz = xv * sh_wih[128 + j] + sh_bih[128 + j];
        const float in = xv * sh_wih[256 + j] + sh_bih[256 + j];
        const float r  = sigmoidf_(ir + hr);
        const float z  = sigmoidf_(iz + hz);
        const float n  = tanhf(in + r * hn);
        const float hp = sh_h[brow][j];
        sh_h[brow][j]  = (1.0f - z) * n + z * hp;
      }
    }
    __syncthreads();
  }

  for (int i = tid; i < 64 * 128; i += 128) {
    const int r = i >> 7, j = i & 127;
    tc[(size_t)(rbase + r) * 128 * 128 + (size_t)f * 128 + j] = sh_h[r][j];
  }
}

// ---------------------------------------------------------------------------
// Kernel 2: transpose LSTM recurrent weights [512,128] -> k-major [128,512]
// so scan B-fragment reads are coalesced b32 row reads.
// ---------------------------------------------------------------------------
__global__ __launch_bounds__(256) void transpose_whh_kernel(
    const float* __restrict__ W, float* __restrict__ Bt) {
  const int idx = blockIdx.x * 256 + threadIdx.x;      // 65536 total
  const int k = idx >> 9, n = idx & 511;
  Bt[idx] = W[n * 128 + k];
}

// ---------------------------------------------------------------------------
// Kernel 3: LSTM recurrent scan (one layer).
// grid = 4 blocks x 32 batch rows, block = 64 (2 waves); wave owns 16 rows.
// Bt: k-major transposed Whh.  pre: [B,S,512] input projections.
// ---------------------------------------------------------------------------
__global__ __launch_bounds__(64) void lstm_scan_kernel(
    const float* __restrict__ pre,
    const float* __restrict__ Bt,    // [128,512]  Bt[k*512+n] = Whh[n][k]
    const float* __restrict__ bhh,   // [512]
    const float* __restrict__ h0,    // [B,128] or nullptr (=> zeros)
    const float* __restrict__ c0,
    float* __restrict__ out,         // [B, S, 128]
    float* __restrict__ hN,          // [B,128]
    float* __restrict__ cN,
    int S)
{
  __shared__ float sh_h[32][HPITCH];
  __shared__ float sh_c[32][HPITCH];
  __shared__ float sh_bhh[512];

  const int rbase = blockIdx.x * 32;
  const int tid   = threadIdx.x;
  const int w     = tid >> 5;
  const int l     = tid & 31;
  const int half  = (l < 16) ? 0 : 1;
  const int l15   = l & 15;

  for (int i = tid; i < 32 * 128; i += 64) {
    const int r = i >> 7, j = i & 127;
    sh_h[r][j] = h0 ? h0[(size_t)(rbase + r) * 128 + j] : 0.0f;
    sh_c[r][j] = c0 ? c0[(size_t)(rbase + r) * 128 + j] : 0.0f;
  }
  for (int i = tid; i < 512; i += 64) sh_bhh[i] = bhh[i];
  __syncthreads();

  const int mrow = w * 16 + l15;

  for (int s = 0; s < S; ++s) {
    const int zo = opaque_zero();

    v8f acc[32];
#pragma unroll
    for (int nt = 0; nt < 32; ++nt) acc[nt] = {};

    for (int k0 = 0; k0 < 128; k0 += 4) {
      const int kk = k0 + 2 * half;
      const v2f a  = *(const v2f*)(&sh_h[mrow][kk]);
      const float* b0 = Bt + (size_t)kk * 512 + zo;   // coalesced rows
      const float* b1 = b0 + 512;
#pragma unroll
      for (int nt = 0; nt < 32; ++nt) {
        v2f b;
        b.x = b0[nt * 16 + l15];
        b.y = b1[nt * 16 + l15];
        acc[nt] = WMMA_F32(a, b, acc[nt]);
      }
    }

#pragma unroll
    for (int jt = 0; jt < 8; ++jt) {
#pragma unroll
      for (int v = 0; v < 8; ++v) {
        const int brow = w * 16 + v + 8 * half;        // local row
        const int grow = rbase + brow;                 // global batch row
        const int j    = jt * 16 + l15;
        const float* prow = pre + ((size_t)grow * S + s) * 512;
        if (jt == 0 && v == 0 && s + 1 < S)
          __builtin_prefetch(prow + 512, 0, 1);        // global_prefetch_b8
        const float gi = acc[jt][v]      + sh_bhh[j]       + prow[j];
        const float gf = acc[8 + jt][v]  + sh_bhh[128 + j] + prow[128 + j];
        const float gg = acc[16 + jt][v] + sh_bhh[256 + j] + prow[256 + j];
        const float go = acc[24 + jt][v] + sh_bhh[384 + j] + prow[384 + j];
        const float ig = sigmoidf_(gi);
        const float fg = sigmoidf_(gf);
        const float g  = tanhf(gg);
        const float og = sigmoidf_(go);
        const float c  = fg * sh_c[brow][j] + ig * g;
        sh_c[brow][j]  = c;
        const float h  = og * tanhf(c);
        sh_h[brow][j]  = h;
        out[((size_t)grow * S + s) * 128 + j] = h;
      }
    }
    __syncthreads();
  }

  for (int i = tid; i < 32 * 128; i += 64) {
    const int r = i >> 7, j = i & 127;
    hN[(size_t)(rbase + r) * 128 + j] = sh_h[r][j];
    cN[(size_t)(rbase + r) * 128 + j] = sh_c[r][j];
  }
}

// ---------------------------------------------------------------------------
// Kernel 4: generic WMMA GEMM  C[M,N] = A[M,K=128] * W[N,K=128]^T + bias[N].
// Block tile 128(M) x 64(N); grid = (M/128, ceil(N/64)); N-guard for N=96.
// ---------------------------------------------------------------------------
__global__ __launch_bounds__(256) void gemm_bias_kernel(
    const float* __restrict__ A, int lda,
    const float* __restrict__ W,
    const float* __restrict__ bias,
    float* __restrict__ C, int ldc, int N)
{
  const int tid  = threadIdx.x;
  const int w    = tid >> 5;
  const int l    = tid & 31;
  const int half = (l < 16) ? 0 : 1;
  const int l15  = l & 15;
  const int mBase = blockIdx.x * 128;
  const int nBase = blockIdx.y * 64;
  const int mrow  = mBase + w * 16 + l15;

  v8f acc[4];
#pragma unroll
  for (int nt = 0; nt < 4; ++nt) acc[nt] = {};

#pragma unroll
  for (int nt = 0; nt < 4; ++nt) {
    const int col   = nBase + nt * 16 + l15;
    const bool okc  = col < N;
    const float* wc = W + (size_t)col * 128;
    for (int k0 = 0; k0 < 128; k0 += 4) {
      v2f a, b;
      a.x = A[(size_t)mrow * lda + k0 + 2 * half];
      a.y = A[(size_t)mrow * lda + k0 + 2 * half + 1];
      b.x = okc ? wc[k0 + 2 * half] : 0.0f;
      b.y = okc ? wc[k0 + 2 * half + 1] : 0.0f;
      acc[nt] = WMMA_F32(a, b, acc[nt]);
    }
  }

#pragma unroll
  for (int nt = 0; nt < 4; ++nt) {
#pragma unroll
    for (int v = 0; v < 8; ++v) {
      const int row = mBase + w * 16 + v + 8 * half;
      const int col = nBase + nt * 16 + l15;
      if (col < N) C[(size_t)row * ldc + col] = acc[nt][v] + bias[col];
    }
  }
}

// ---------------------------------------------------------------------------
// Kernel 5: MHA scores/softmax/context (tiny, VALU).  grid = 128 (query n),
// block = 128 = 4 wave32s, one head per wave.  Reproduces the reference's
// exact (transposed) indexing: scores[n,h,s] = qp[n,h,:] . KP[s*128+n, h,:].
// ---------------------------------------------------------------------------
__global__ __launch_bounds__(128) void attn_kernel(
    const float* __restrict__ qp,   // [128,128]
    const float* __restrict__ KP,   // [16384,128]
    const float* __restrict__ VP,   // [16384,128]
    float* __restrict__ ctx)        // [128,128]
{
  __shared__ float sq[128];
  __shared__ float satt[4][128];
  const int n = blockIdx.x;
  const int tid = threadIdx.x;
  const int h = tid >> 5;
  const int l = tid & 31;

  sq[tid] = qp[n * 128 + tid];
  __syncthreads();

  const float scale = 0.17677669529663687f;  // 1/sqrt(DH=32)
  float sc[4];
#pragma unroll
  for (int si = 0; si < 4; ++si) {
    const int s = l + 32 * si;
    const float* kr = KP + ((size_t)s * 128 + n) * 128 + h * 32;
    float a = 0.0f;
    for (int d = 0; d < 32; ++d) a += sq[h * 32 + d] * kr[d];
    sc[si] = a * scale;
  }
  float m = fmaxf(fmaxf(sc[0], sc[1]), fmaxf(sc[2], sc[3]));
  for (int off = 16; off > 0; off >>= 1) m = fmaxf(m, __shfl_xor(m, off, 32));
  float e[4], sum = 0.0f;
#pragma unroll
  for (int si = 0; si < 4; ++si) { e[si] = __expf(sc[si] - m); sum += e[si]; }
  for (int off = 16; off > 0; off >>= 1) sum += __shfl_xor(sum, off, 32);
  const float inv = 1.0f / sum;
#pragma unroll
  for (int si = 0; si < 4; ++si) satt[h][l + 32 * si] = e[si] * inv;
  __syncthreads();

  float ca = 0.0f;  // lane = output dim d
  for (int s = 0; s < 128; ++s)
    ca += satt[h][s] * VP[((size_t)s * 128 + n) * 128 + h * 32 + l];
  ctx[n * 128 + h * 32 + l] = ca;
}

// ---------------------------------------------------------------------------
extern "C" void kernel_launch(void* const* d_in, const int* in_sizes, int n_in,
                              void* d_out, int out_size, void* d_ws, size_t ws_size,
                              hipStream_t stream) {
  const float* x        = (const float*)d_in[0];
  const float* gWih     = (const float*)d_in[1];
  const float* gWhh     = (const float*)d_in[2];
  const float* gbih     = (const float*)d_in[3];
  const float* gbhh     = (const float*)d_in[4];
  const float* encWih   = (const float*)d_in[5];   // [2,512,128]
  const float* encWhh   = (const float*)d_in[6];
  const float* encbih   = (const float*)d_in[7];   // [2,512]
  const float* encbhh   = (const float*)d_in[8];
  const float* decWih   = (const float*)d_in[9];
  const float* decWhh   = (const float*)d_in[10];
  const float* decbih   = (const float*)d_in[11];
  const float* decbhh   = (const float*)d_in[12];
  const float* ainw     = (const float*)d_in[13];  // [384,128]
  const float* ainb     = (const float*)d_in[14];  // [384]
  const float* aoutw    = (const float*)d_in[15];  // [128,128]
  const float* aoutb    = (const float*)d_in[16];  // [128]
  const float* projW    = (const float*)d_in[17];  // [96,128]
  const float* projb    = (const float*)d_in[18];  // [96]
  float* out = (float*)d_out;

  float* ws = (float*)d_ws;
  float* tc    = ws;                    // [128,128,128] temporal_concat
  float* pre   = tc + 2097152;          // [16384,512]   LSTM input projections
  float* enc0  = pre + 8388608;         // [128,128,128]
  float* enc1  = enc0 + 2097152;        // [128,128,128]
  float* hn    = enc1 + 2097152;        // [2,128,128]
  float* cn    = hn + 32768;
  float* dpre  = cn + 32768;            // [128,512]
  float* dec0  = dpre + 65536;          // [128,128]
  float* dout  = dec0 + 16384;          // [128,128]
  float* junkH = dout + 16384;
  float* junkC = junkH + 16384;
  float* qp    = junkC + 16384;
  float* ctx   = qp + 16384;
  float* attd  = ctx + 16384;
  float* BtE0  = attd + 16384;          // [128,512] transposed Whh x4
  float* BtE1  = BtE0 + 65536;
  float* BtD0  = BtE1 + 65536;
  float* BtD1  = BtD0 + 65536;
  float* KP    = pre;                   // reuse pre region after decoder
  float* VP    = pre + 2097152;

  // 0) transpose recurrent weights to k-major (coalesced scan reads)
  transpose_whh_kernel<<<256, 256, 0, stream>>>(encWhh, BtE0);
  transpose_whh_kernel<<<256, 256, 0, stream>>>(encWhh + 512 * 128, BtE1);
  transpose_whh_kernel<<<256, 256, 0, stream>>>(decWhh, BtD0);
  transpose_whh_kernel<<<256, 256, 0, stream>>>(decWhh + 512 * 128, BtD1);

  // 1) per-feature GRU scan  (dominant: ~412 GFLOP of f32 WMMA)
  gru_featurewise_kernel<<<256, 128, 0, stream>>>(x, gWih, gWhh, gbih, gbhh, tc);

  // 2) encoder LSTM
  gemm_bias_kernel<<<dim3(128, 8), 256, 0, stream>>>(tc, 128, encWih, encbih, pre, 512, 512);
  lstm_scan_kernel<<<4, 64, 0, stream>>>(pre, BtE0, encbhh, nullptr, nullptr,
                                         enc0, hn, cn, 128);
  gemm_bias_kernel<<<dim3(128, 8), 256, 0, stream>>>(enc0, 128, encWih + 512 * 128,
                                                     encbih + 512, pre, 512, 512);
  lstm_scan_kernel<<<4, 64, 0, stream>>>(pre, BtE1, encbhh + 512, nullptr, nullptr,
                                         enc1, hn + 16384, cn + 16384, 128);

  // 3) decoder LSTM on last timestep, seeded with (h_n, c_n)
  gemm_bias_kernel<<<dim3(1, 8), 256, 0, stream>>>(tc + 127 * 128, 16384, decWih,
                                                   decbih, dpre, 512, 512);
  lstm_scan_kernel<<<4, 64, 0, stream>>>(dpre, BtD0, decbhh, hn, cn,
                                         dec0, junkH, junkC, 1);
  gemm_bias_kernel<<<dim3(1, 8), 256, 0, stream>>>(dec0, 128, decWih + 512 * 128,
                                                   decbih + 512, dpre, 512, 512);
  lstm_scan_kernel<<<4, 64, 0, stream>>>(dpre, BtD1, decbhh + 512,
                                         hn + 16384, cn + 16384, dout, junkH, junkC, 1);

  // 4) attention projections
  gemm_bias_kernel<<<dim3(1, 2), 256, 0, stream>>>(dout, 128, ainw, ainb, qp, 128, 128);
  gemm_bias_kernel<<<dim3(128, 2), 256, 0, stream>>>(enc1, 128, ainw + 128 * 128,
                                                     ainb + 128, KP, 128, 128);
  gemm_bias_kernel<<<dim3(128, 2), 256, 0, stream>>>(enc1, 128, ainw + 2 * 128 * 128,
                                                     ainb + 256, VP, 128, 128);

  // 5) softmax attention + context
  attn_kernel<<<128, 128, 0, stream>>>(qp, KP, VP, ctx);

  // 6) output projection, then final projection -> [128, 24, 4]
  gemm_bias_kernel<<<dim3(1, 2), 256, 0, stream>>>(ctx, 128, aoutw, aoutb, attd, 128, 128);
  gemm_bias_kernel<<<dim3(1, 2), 256, 0, stream>>>(attd, 128, projW, projb, out, 96, 96);
}